// enhancedEMPSN_36309653520608
// MI455X (gfx1250) — compile-verified
//
#include <hip/hip_runtime.h>
#include <math.h>

// ---------------------------------------------------------------------------
// EMPSN-style simplicial message passing network for gfx1250 (MI455X).
// fp32 end-to-end using V_WMMA_F32_16X16X4_F32 (wave32, 16x16 C/D tiles),
// with the weight panel staged in LDS via GLOBAL_LOAD_ASYNC_TO_LDS_B128.
// ---------------------------------------------------------------------------

typedef float v2f __attribute__((ext_vector_type(2)));
typedef float v8f __attribute__((ext_vector_type(8)));

#define HDIM 128

enum { ACT_NONE = 0, ACT_SIGMOID = 1, ACT_SILU = 2 };

__device__ __forceinline__ float sigmoidf_(float x) {
  return 1.0f / (1.0f + __expf(-x));
}

// Stage `nbytes` of B (row-major [K,128] panel) into LDS with the async copy
// engine: each lane moves 16B per op, tracked by ASYNCcnt. nbytes must be a
// multiple of 4096 (256 threads * 16B) -- true for K in {32,128}.
__device__ __forceinline__ void stage_b_async(const float* __restrict__ B,
                                              float* bsh, int nbytes) {
  const int tid = threadIdx.x;
  for (int ofs = tid * 16; ofs < nbytes; ofs += 256 * 16) {
    unsigned lds_a = (unsigned)(size_t)((char*)bsh + ofs);             // LDS byte offset
    unsigned long long ga = (unsigned long long)(size_t)((const char*)B + ofs);
    asm volatile("global_load_async_to_lds_b128 %0, %1, off"
                 :: "v"(lds_a), "v"(ga) : "memory");
  }
  asm volatile("s_wait_asynccnt 0" ::: "memory");
  __syncthreads();
}

// ---------------------------------------------------------------------------
// D[M,128] = act(A[M,K] @ B[K,128] (+ bias))
// grid.x = M/64 (exact), block = 256 (8 waves).
// Block covers 64 rows x 128 cols; wave w: rows r0 = 64*bx + 16*(w>>1),
// cols [64*(w&1), 64*(w&1)+64) as four 16x16 WMMA tiles -> one A fragment
// feeds 4 WMMAs; B fragments come from LDS (ds_load pair, read2-fusable).
// A fragment (16x4 f32): lane<16 -> (K,K+1) of row (lane&15); lane>=16 -> (K+2,K+3).
// C/D tile: acc[v] -> row r0 + v + 8*(lane>=16), col c0 + (lane&15).
// ---------------------------------------------------------------------------
template <int ACT, bool BIAS>
__global__ void gemm128_wmma(const float* __restrict__ A,
                             const float* __restrict__ B,
                             const float* __restrict__ bias,
                             float* __restrict__ D, int K) {
  __shared__ __align__(16) float bsh[HDIM * HDIM];  // 64 KB (K<=128 rows used)
  stage_b_async(B, bsh, K * HDIM * 4);

  const int wave = threadIdx.x >> 5;
  const int lane = threadIdx.x & 31;
  const int hi = lane >> 4;      // which 16-lane half
  const int lo = lane & 15;
  const size_t r0 = (size_t)blockIdx.x * 64 + 16 * (wave >> 1);
  const int ch = 64 * (wave & 1);

  const float* arow = A + (r0 + (size_t)lo) * (size_t)K + 2 * hi;
  const float* brow = bsh + (size_t)(2 * hi) * HDIM + ch + lo;

  v8f acc[4] = {};
  for (int k = 0; k < K; k += 4) {
    v2f a = *(const v2f*)(arow + k);           // 8B aligned: k%4==0, strides even
    const float* bk = brow + (size_t)k * HDIM;
#pragma unroll
    for (int t = 0; t < 4; ++t) {
      v2f b;
      b.x = bk[t * 16];                        // bsh[k+2hi  ][ch+16t+lo]
      b.y = bk[t * 16 + HDIM];                 // bsh[k+2hi+1][ch+16t+lo]
      acc[t] = __builtin_amdgcn_wmma_f32_16x16x4_f32(
          false, a, false, b, (short)0, acc[t], false, false);
    }
  }

  float* drow = D + (r0 + (size_t)(8 * hi)) * HDIM + ch + lo;
#pragma unroll
  for (int t = 0; t < 4; ++t) {
    const float bv = BIAS ? bias[ch + 16 * t + lo] : 0.0f;
#pragma unroll
    for (int v = 0; v < 8; ++v) {
      float val = acc[t][v];
      if (BIAS) val += bv;
      if (ACT == ACT_SIGMOID) val = sigmoidf_(val);
      else if (ACT == ACT_SILU) val = val * sigmoidf_(val);
      drow[(size_t)v * HDIM + 16 * t] = val;
    }
  }
}

// ---------------------------------------------------------------------------
// Fused: tmp = A[M,128] @ B[128,128] + bias;  p[batch[row]*384 + rsec*128+col] += tmp
// (second pre-pool linear fused with global_add_pool; writes straight into the
//  concatenated [G,384] state buffer -> no extra [N,128] tensor, no concat pass)
// ---------------------------------------------------------------------------
__global__ void gemm128_pool_wmma(const float* __restrict__ A,
                                  const float* __restrict__ B,
                                  const float* __restrict__ bias,
                                  const int* __restrict__ batch,
                                  float* __restrict__ p, int rsec) {
  __shared__ __align__(16) float bsh[HDIM * HDIM];  // 64 KB
  stage_b_async(B, bsh, HDIM * HDIM * 4);

  const int wave = threadIdx.x >> 5;
  const int lane = threadIdx.x & 31;
  const int hi = lane >> 4;
  const int lo = lane & 15;
  const size_t r0 = (size_t)blockIdx.x * 64 + 16 * (wave >> 1);
  const int ch = 64 * (wave & 1);

  const float* arow = A + (r0 + (size_t)lo) * HDIM + 2 * hi;
  const float* brow = bsh + (size_t)(2 * hi) * HDIM + ch + lo;

  v8f acc[4] = {};
  for (int k = 0; k < HDIM; k += 4) {
    v2f a = *(const v2f*)(arow + k);
    const float* bk = brow + (size_t)k * HDIM;
#pragma unroll
    for (int t = 0; t < 4; ++t) {
      v2f b;
      b.x = bk[t * 16];
      b.y = bk[t * 16 + HDIM];
      acc[t] = __builtin_amdgcn_wmma_f32_16x16x4_f32(
          false, a, false, b, (short)0, acc[t], false, false);
    }
  }

#pragma unroll
  for (int t = 0; t < 4; ++t) {
    const int col = ch + 16 * t + lo;
    const float bv = bias[col];
#pragma unroll
    for (int v = 0; v < 8; ++v) {
      const int row = (int)r0 + 8 * hi + v;
      const int g = batch[row];
      atomicAdd(&p[(size_t)g * 384 + rsec * HDIM + col], acc[t][v] + bv);
    }
  }
}

// ---------------------------------------------------------------------------
// m[cols[e]] += h[rows[e]]  over 128-wide rows; one wave per edge, float4/lane.
// ---------------------------------------------------------------------------
__global__ void scatter_add_seg(const float* __restrict__ h,
                                const int* __restrict__ rows,
                                const int* __restrict__ cols,
                                float* __restrict__ m, int nnz) {
  const int e = blockIdx.x * 8 + (threadIdx.x >> 5);
  if (e >= nnz) return;
  const int lane = threadIdx.x & 31;
  const int r = rows[e];
  const int c = cols[e];
  const float4 v = *(const float4*)(h + (size_t)r * HDIM + lane * 4);
  float* dst = m + (size_t)c * HDIM + lane * 4;
  atomicAdd(dst + 0, v.x);
  atomicAdd(dst + 1, v.y);
  atomicAdd(dst + 2, v.z);
  atomicAdd(dst + 3, v.w);
}

__global__ void sigmoid_inplace(float* __restrict__ h, long n) {
  const long i = (long)blockIdx.x * blockDim.x + threadIdx.x;
  if (i < n) h[i] = sigmoidf_(h[i]);
}

// ---------------------------------------------------------------------------
// out[g] = silu(p[g,:384] @ post_w1 + post_b1) @ post_w2 + post_b2
// one block (128 threads) per graph; LDS stage + LDS tree reduction.
// ---------------------------------------------------------------------------
__global__ void final_mlp(const float* __restrict__ p,
                          const float* __restrict__ w1,
                          const float* __restrict__ b1,
                          const float* __restrict__ w2,
                          const float* __restrict__ b2,
                          float* __restrict__ out) {
  __shared__ float srow[384];
  __shared__ float red[128];
  const int g = blockIdx.x;
  const int t = threadIdx.x;  // 128 threads
  for (int k = t; k < 384; k += 128) srow[k] = p[(size_t)g * 384 + k];
  __syncthreads();
  float acc = b1[t];
  for (int k = 0; k < 384; ++k) acc = fmaf(srow[k], w1[(size_t)k * HDIM + t], acc);
  const float z = acc * sigmoidf_(acc);
  red[t] = z * w2[t];
  __syncthreads();
  for (int s = 64; s > 0; s >>= 1) {
    if (t < s) red[t] += red[t + s];
    __syncthreads();
  }
  if (t == 0) out[g] = red[0] + b2[0];
}

// ---------------------------------------------------------------------------
extern "C" void kernel_launch(void* const* d_in, const int* in_sizes, int n_in,
                              void* d_out, int out_size, void* d_ws, size_t ws_size,
                              hipStream_t stream) {
  constexpr int N0 = 200000, N1 = 600000, N2 = 400000;
  constexpr int NNZ1 = 1200000, NNZ2 = 1200000;
  constexpr int G = 256, NIN = 32, L = 2;
  constexpr int HH = HDIM * HDIM;

  const float* x0      = (const float*)d_in[0];
  const float* x1      = (const float*)d_in[1];
  const float* x2      = (const float*)d_in[2];
  const int*   b1_rows = (const int*)d_in[3];
  const int*   b1_cols = (const int*)d_in[4];
  const int*   b2_rows = (const int*)d_in[5];
  const int*   b2_cols = (const int*)d_in[6];
  const int*   batch0  = (const int*)d_in[7];
  const int*   batch1  = (const int*)d_in[8];
  const int*   batch2  = (const int*)d_in[9];
  const float* w_emb   = (const float*)d_in[10];
  const float* b_emb   = (const float*)d_in[11];
  const float* conv_w  = (const float*)d_in[12];  // [L,2,H,H]
  const float* pre_w1  = (const float*)d_in[13];  // [3,H,H]
  const float* pre_b1  = (const float*)d_in[14];  // [3,H]
  const float* pre_w2  = (const float*)d_in[15];  // [3,H,H]
  const float* pre_b2  = (const float*)d_in[16];  // [3,H]
  const float* post_w1 = (const float*)d_in[17];  // [384,H]
  const float* post_b1 = (const float*)d_in[18];  // [H]
  const float* post_w2 = (const float*)d_in[19];  // [H,1]
  const float* post_b2 = (const float*)d_in[20];  // [1]
  float* out = (float*)d_out;

  // ---- workspace carve-up (256B aligned) ----
  size_t off = 0;
  auto carve = [&](size_t bytes) -> float* {
    float* ptr = (float*)((char*)d_ws + off);
    off += (bytes + 255) & ~(size_t)255;
    return ptr;
  };
  float* h0 = carve((size_t)N0 * HDIM * 4);
  float* h1 = carve((size_t)N1 * HDIM * 4);
  float* h2 = carve((size_t)N2 * HDIM * 4);
  float* m1 = carve((size_t)N1 * HDIM * 4);  // also reused as z-buffer (max N1 rows)
  float* m2 = carve((size_t)N2 * HDIM * 4);
  float* p  = carve((size_t)G * 384 * 4);    // concatenated [G, 3H] pooled state
  (void)ws_size; (void)n_in; (void)in_sizes; (void)out_size;

  const dim3 blk(256);

  // ---- embedding: h_r = x_r @ w_emb + b_emb ----
  gemm128_wmma<ACT_NONE, true><<<dim3(N0 / 64), blk, 0, stream>>>(x0, w_emb, b_emb, h0, NIN);
  gemm128_wmma<ACT_NONE, true><<<dim3(N1 / 64), blk, 0, stream>>>(x1, w_emb, b_emb, h1, NIN);
  gemm128_wmma<ACT_NONE, true><<<dim3(N2 / 64), blk, 0, stream>>>(x2, w_emb, b_emb, h2, NIN);

  // ---- message passing layers ----
  for (int l = 0; l < L; ++l) {
    hipMemsetAsync(m1, 0, (size_t)N1 * HDIM * 4, stream);
    hipMemsetAsync(m2, 0, (size_t)N2 * HDIM * 4, stream);
    scatter_add_seg<<<dim3((NNZ1 + 7) / 8), blk, 0, stream>>>(h0, b1_rows, b1_cols, m1, NNZ1);
    scatter_add_seg<<<dim3((NNZ2 + 7) / 8), blk, 0, stream>>>(h1, b2_rows, b2_cols, m2, NNZ2);
    sigmoid_inplace<<<dim3((N0 * HDIM) / 256), blk, 0, stream>>>(h0, (long)N0 * HDIM);
    gemm128_wmma<ACT_SIGMOID, false><<<dim3(N1 / 64), blk, 0, stream>>>(
        m1, conv_w + (size_t)(l * 2 + 0) * HH, nullptr, h1, HDIM);
    gemm128_wmma<ACT_SIGMOID, false><<<dim3(N2 / 64), blk, 0, stream>>>(
        m2, conv_w + (size_t)(l * 2 + 1) * HH, nullptr, h2, HDIM);
  }

  // ---- per-rank pre-pool MLP + fused global_add_pool into p[G,384] ----
  hipMemsetAsync(p, 0, (size_t)G * 384 * 4, stream);

  const float* hs[3]    = {h0, h1, h2};
  const int    ns[3]    = {N0, N1, N2};
  const int*   bats[3]  = {batch0, batch1, batch2};
  for (int r = 0; r < 3; ++r) {
    // z = silu(h @ pre_w1[r] + pre_b1[r])   (z lives in m1 scratch)
    gemm128_wmma<ACT_SILU, true><<<dim3(ns[r] / 64), blk, 0, stream>>>(
        hs[r], pre_w1 + (size_t)r * HH, pre_b1 + (size_t)r * HDIM, m1, HDIM);
    // p[batch,:][r-section] += z @ pre_w2[r] + pre_b2[r]
    gemm128_pool_wmma<<<dim3(ns[r] / 64), blk, 0, stream>>>(
        m1, pre_w2 + (size_t)r * HH, pre_b2 + (size_t)r * HDIM, bats[r], p, r);
  }

  // ---- head: out = silu(p @ post_w1 + post_b1) @ post_w2 + post_b2 ----
  final_mlp<<<dim3(G), dim3(128), 0, stream>>>(p, post_w1, post_b1, post_w2, post_b2, out);
}